// AdaConv_73023033967446
// MI455X (gfx1250) — compile-verified
//
#include <hip/hip_runtime.h>
#include <hip/hip_bf16.h>

typedef __attribute__((ext_vector_type(2))) float v2f;
typedef __attribute__((ext_vector_type(8))) float v8f;

#define BB   4
#define CC   64
#define OO   64
#define HH   128
#define WW   128
#define HXW  (HH * WW)          // 16384
#define K2   9                  // 3x3 taps
#define KTOT (CC * K2)          // 576 contraction length
#define NPIX 32                 // pixels per workgroup (two WMMA N tiles)
#define BSTR 40                 // LDS row stride (floats): K-halves -> disjoint banks

// ---------------------------------------------------------------------------
// Phase 1: per pixel, select 9 smallest of the 25 window distances (stable,
// ascending -> matches jax.lax.top_k on -l), emit reflect-padded gather
// offsets (u16 index into the HxW plane), 10 u16 per pixel (last is pad).
// ---------------------------------------------------------------------------
__global__ __launch_bounds__(256) void adaconv_select(
    const float* __restrict__ l, unsigned short* __restrict__ offs)
{
  int pix = blockIdx.x * 256 + threadIdx.x;       // 0 .. B*HXW-1
  int b = pix >> 14;
  int hw = pix & (HXW - 1);
  int h = hw >> 7;
  int w = hw & (WW - 1);
  const float* lb = l + b * HXW;

  float v[25];
#pragma unroll
  for (int dy = -2; dy <= 2; ++dy) {
#pragma unroll
    for (int dx = -2; dx <= 2; ++dx) {
      int i = (dy + 2) * 5 + (dx + 2);
      int y = h + dy, xq = w + dx;
      bool ok = (y >= 0) && (y < HH) && (xq >= 0) && (xq < WW);
      v[i] = ok ? lb[y * WW + xq] : 999.0f;       // constant-pad 999
    }
  }

  unsigned int taken = 0;
  size_t base = (size_t)pix * 10;
#pragma unroll
  for (int j = 0; j < K2; ++j) {
    float best = 1e30f;
    int bi = 0;
#pragma unroll
    for (int i = 0; i < 25; ++i) {
      bool better = (((taken >> i) & 1u) == 0u) && (v[i] < best);
      best = better ? v[i] : best;
      bi = better ? i : bi;
    }
    taken |= (1u << bi);
    int dy = bi / 5 - 2, dx = bi % 5 - 2;
    int sy = h + dy; sy = sy < 0 ? -sy : (sy > HH - 1 ? 2 * HH - 2 - sy : sy);
    int sx = w + dx; sx = sx < 0 ? -sx : (sx > WW - 1 ? 2 * WW - 2 - sx : sx);
    offs[base + j] = (unsigned short)(sy * WW + sx);
  }
  offs[base + 9] = 0;
}

// ---------------------------------------------------------------------------
// Phase 2: out[b, 0..63, n0..n0+31] via V_WMMA_F32_16X16X4_F32.
// 4 waves; wave w handles output rows o = 16w..16w+15 for BOTH 16-pixel
// N-subtiles (A fragment loaded once, used twice -> halves weight traffic).
// Gather fill uses GLOBAL_LOAD_ASYNC_TO_LDS_B32 (ASYNCcnt-tracked).
// ---------------------------------------------------------------------------
__global__ __launch_bounds__(128) void adaconv_wmma(
    const float* __restrict__ x, const float* __restrict__ weight,
    const float* __restrict__ bias, const unsigned short* __restrict__ offs,
    float* __restrict__ out)
{
  __shared__ float Bt[KTOT * BSTR];               // 92160 B
  __shared__ unsigned short offl[NPIX * 10];      // 640 B

  int wg = blockIdx.x;                            // 0 .. B*HXW/NPIX - 1
  int b  = wg >> 9;                               // HXW/NPIX = 512 strips/batch
  int n0 = (wg & 511) * NPIX;

  int t = threadIdx.x;
  int lane = t & 31;
  int wave = t >> 5;

  // stage gather offsets for the 32 pixels of this strip
  for (int q = t; q < NPIX * K2; q += 128) {
    int n = q / K2, j = q - n * K2;
    offl[n * 10 + j] = offs[(size_t)(b * HXW + n0 + n) * 10 + j];
  }
  __syncthreads();

  // cooperative gather straight into LDS: Bt[kk][n] = x[b, c, off(n,j)]
  // kk = c*9 + j.  Async global->LDS, per-lane addresses, ASYNCcnt-tracked.
  const float* xb = x + (size_t)b * CC * HXW;
  for (int e = t; e < KTOT * NPIX; e += 128) {    // exactly 144 full-wave iters
    int kk = e >> 5;                              // NPIX = 32
    int n  = e & 31;
    int c  = kk / K2;
    int j  = kk - c * K2;
    int off = offl[n * 10 + j];
    const float* gsrc = xb + c * HXW + off;
    unsigned ldsdst = (unsigned)(size_t)&Bt[kk * BSTR + n];
    asm volatile("global_load_async_to_lds_b32 %0, %1, off"
                 :: "v"(ldsdst), "v"(gsrc) : "memory");
  }
  asm volatile("s_wait_asynccnt 0x0" ::: "memory");
  __syncthreads();

  // accumulators (two N-subtiles) init with bias, per 16x16 f32 C/D layout:
  // VGPR r: lanes 0-15 -> M=r, lanes 16-31 -> M=r+8; N = lane & 15.
  int mhalf = (lane >> 4) << 3;                   // 0 or 8
  int nloc  = lane & 15;
  int obase = wave * 16;
  v8f acc0, acc1;
#pragma unroll
  for (int r = 0; r < 8; ++r) { acc0[r] = bias[obase + r + mhalf]; acc1[r] = acc0[r]; }

  // A layout (16x4 f32): lanes 0-15 hold (M=lane, K=k0 / k0+1),
  // lanes 16-31 hold (M=lane-16, K=k0+2 / k0+3).  weight is O x 576 row-major
  // with flat K = c*9 + j, matching Bt's kk ordering.
  int kshift = (lane >> 4) << 1;                  // 0 or 2
  const float* wrow = weight + (size_t)(obase + (lane & 15)) * KTOT + kshift;

  for (int kk0 = 0; kk0 < KTOT; kk0 += 4) {
    v2f a = *(const v2f*)(wrow + kk0);            // global_load_b64, L2-resident
    int kb = kk0 + kshift;
    v2f b0, b1;
    b0.x = Bt[kb * BSTR + nloc];
    b0.y = Bt[(kb + 1) * BSTR + nloc];
    b1.x = Bt[kb * BSTR + 16 + nloc];
    b1.y = Bt[(kb + 1) * BSTR + 16 + nloc];
    acc0 = __builtin_amdgcn_wmma_f32_16x16x4_f32(false, a, false, b0,
                                                 (short)0, acc0, false, false);
    acc1 = __builtin_amdgcn_wmma_f32_16x16x4_f32(false, a, false, b1,
                                                 (short)0, acc1, false, false);
  }

  // store D per C/D layout (both subtiles)
  float* ob = out + (size_t)b * OO * HXW + n0 + nloc;
#pragma unroll
  for (int r = 0; r < 8; ++r) {
    size_t row = (size_t)(obase + r + mhalf) * HXW;
    ob[row]      = acc0[r];
    ob[row + 16] = acc1[r];
  }
}

extern "C" void kernel_launch(void* const* d_in, const int* in_sizes, int n_in,
                              void* d_out, int out_size, void* d_ws, size_t ws_size,
                              hipStream_t stream) {
  const float* x      = (const float*)d_in[0];
  const float* l      = (const float*)d_in[1];
  const float* weight = (const float*)d_in[2];
  const float* bias   = (const float*)d_in[3];
  // d_in[4] = kernel_size (3), d_in[5] = window_size (5): compile-time constants here
  unsigned short* offs = (unsigned short*)d_ws;   // B*HXW*10 u16 = 1.31 MB
  float* out = (float*)d_out;

  adaconv_select<<<(BB * HXW) / 256, 256, 0, stream>>>(l, offs);
  adaconv_wmma<<<BB * (HXW / NPIX), 128, 0, stream>>>(x, weight, bias, offs, out);
}